// Predictor_30073361006778
// MI455X (gfx1250) — compile-verified
//
#include <hip/hip_runtime.h>

// Problem constants (from reference): B=256, T=256, E=64, H=4, fp32 throughout.
#define BB 256
#define TT 256
#define EE 64
#define HH 4

typedef __attribute__((ext_vector_type(2))) float v2f;
typedef __attribute__((ext_vector_type(4))) float v4f;
typedef __attribute__((ext_vector_type(8))) float v8f;

// CDNA5 fp32 matrix op: D(16x16) = A(16x4) * B(4x16) + C.
// A frag: lanes 0-15 = M rows, VGPR0/1 = K{0,1}; lanes 16-31 same rows, K{2,3}.
// B frag: lanes = N cols, VGPR0/1 = K rows (half-wave selects K pair), mirrored.
// C/D  : VGPR r -> M = r (lanes 0-15) / r+8 (lanes 16-31), N = lane%16.
static __device__ __forceinline__ v8f wmma_f32_16x16x4(v2f a, v2f b, v8f c) {
  return __builtin_amdgcn_wmma_f32_16x16x4_f32(
      /*neg_a=*/false, a, /*neg_b=*/false, b,
      /*c_mod=*/(short)0, c, /*reuse_a=*/false, /*reuse_b=*/false);
}

// B-matrix LDS layout: K-pair interleaved so one fragment = one aligned v2f load:
//   Bp[(k>>1)*BSTRIDE + n*2 + (k&1)]  holds B[k][n].
// BSTRIDE=160 (mod 64 == 32) puts the two half-waves (pair-rows p and p+1) on
// disjoint bank sets -> conflict-free ds_load_b64 fragment reads.
#define BSTRIDE 160
// A-matrix LDS stride 68: row-to-row bank shift of 4 -> conflict-free b64 reads.
#define ASTRIDE 68

// ---------------------------------------------------------------------------
// Kernel 1: for fixed j, h[:, :, j, :] = relu(x0[:,j,:] @ W1[:,j,:,:] + b1).
// GEMM view: M = batch rows, K = E = 64, N = (i,h) columns -> full 16-wide
// WMMA tiles (the naive per-(i,j) layout would waste 75% of N).
// Diagonal (i==j) entries forced to 0 so kernel 2 can run dense K=1024.
// h_ws layout: (i, b, j, h) -> stage-2 A rows contiguous over (j,h).
// Grid: (j=256, iblock=16, bhalf=2). Block: 128 threads = 4 waves.
// Each wave: 2 M-tiles x 4 N-tiles -> 8 WMMAs per 6 LDS fragment loads.
// ---------------------------------------------------------------------------
__global__ __launch_bounds__(128) void k1_h(
    const float* __restrict__ x,    // (B,1,T,E)
    const float* __restrict__ W1,   // (T,T,E,H)
    const float* __restrict__ b1,   // (T,T,H)
    float* __restrict__ h_ws) {     // (T_i, B, T_j, H)
  const int j  = blockIdx.x;
  const int ib = blockIdx.y;        // i in [ib*16, ib*16+16)
  const int mh = blockIdx.z;        // b in [mh*128, mh*128+128)
  const int tid = threadIdx.x;
  const int wave = tid >> 5, lane = tid & 31;
  const int half = lane >> 4, lm = lane & 15;

  __shared__ float Axs[128 * ASTRIDE];  // x0 tile (128 x 64)
  __shared__ float Bxs[32 * BSTRIDE];   // W1 tile, K-pair interleaved (64 x 64)

  // Stage A: float4-coalesced (16 lanes span one 256B row of x).
  for (int idx = tid; idx < 128 * 16; idx += 128) {
    int bl = idx >> 4, e4 = idx & 15;
    v4f v = *(const v4f*)(x + (size_t)(mh * 128 + bl) * (TT * EE) + (size_t)j * EE + e4 * 4);
    *(v4f*)(Axs + bl * ASTRIDE + e4 * 4) = v;
  }
  // Stage B: each i's W1[i,j] slab is 1KB contiguous (e-major, 4 h per float4);
  // scatter into K-pair-interleaved layout (n = il*4+h).
  for (int idx = tid; idx < 16 * 64; idx += 128) {
    int il = idx >> 6, e = idx & 63;
    int i = ib * 16 + il;
    v4f v = *(const v4f*)(W1 + ((size_t)i * TT + j) * (EE * HH) + e * 4);
    float* dst = Bxs + (e >> 1) * BSTRIDE + (e & 1);
    int n0 = il * 4;
    dst[(n0 + 0) * 2] = v.x;
    dst[(n0 + 1) * 2] = v.y;
    dst[(n0 + 2) * 2] = v.z;
    dst[(n0 + 3) * 2] = v.w;
  }
  __syncthreads();

  v8f acc[2][4] = {};               // 2 M-tiles x 4 N-tiles per wave (32x64 out)
  const int m0 = wave * 32;
  for (int e = 0; e < EE; e += 4) {
    v2f a0 = *(const v2f*)(Axs + (m0 + lm) * ASTRIDE + e + half * 2);
    v2f a1 = *(const v2f*)(Axs + (m0 + 16 + lm) * ASTRIDE + e + half * 2);
#pragma unroll
    for (int n = 0; n < 4; ++n) {
      v2f bf = *(const v2f*)(Bxs + ((e >> 1) + half) * BSTRIDE + (n * 16 + lm) * 2);
      acc[0][n] = wmma_f32_16x16x4(a0, bf, acc[0][n]);
      acc[1][n] = wmma_f32_16x16x4(a1, bf, acc[1][n]);
    }
  }

  // Epilogue: +b1, relu, zero the i==j diagonal, scatter to h_ws.
#pragma unroll
  for (int mt = 0; mt < 2; ++mt) {
#pragma unroll
    for (int n = 0; n < 4; ++n) {
      int ncol = n * 16 + lm;
      int i = ib * 16 + (ncol >> 2);
      int hh = ncol & 3;
      float bias = b1[((size_t)i * TT + j) * HH + hh];
      float keep = (i == j) ? 0.f : 1.f;
#pragma unroll
      for (int r = 0; r < 8; ++r) {
        int brow = mh * 128 + m0 + mt * 16 + r + half * 8;
        float v = acc[mt][n][r] + bias;
        v = v > 0.f ? v : 0.f;
        h_ws[(((size_t)i * BB + brow) * TT + j) * HH + hh] = v * keep;
      }
    }
  }
}

// ---------------------------------------------------------------------------
// Kernel 2: out[:, t, :] = h_ws[t] @ W2'[t] + b2[t]. M=128 (b half), N=64 (e),
// K=1024 dense over (j,h); the off-diagonal pack is the closed-form remap
// kw = (j<t ? k : j>t ? k-4 : 0) on W2 rows (j==t rows multiply h==0).
// Grid: (t=256, bhalf=2). Block: 128 threads = 4 waves, 2Mx4N tiles per wave.
// ---------------------------------------------------------------------------
__global__ __launch_bounds__(128) void k2_out(
    const float* __restrict__ h_ws, // (T, B, T*H)
    const float* __restrict__ W2,   // (T, (T-1)*H, E)
    const float* __restrict__ b2,   // (T, E)
    float* __restrict__ out) {      // (B, T, E)
  const int t = blockIdx.x;
  const int mh = blockIdx.y;
  const int tid = threadIdx.x;
  const int wave = tid >> 5, lane = tid & 31;
  const int half = lane >> 4, lm = lane & 15;

  __shared__ float As[128 * ASTRIDE];   // h chunk (128 b x 64 k)
  __shared__ float Bs[32 * BSTRIDE];    // W2 chunk, K-pair interleaved (64 k x 64 e)

  v8f acc[2][4] = {};
  const int m0 = wave * 32;

  for (int k0 = 0; k0 < TT * HH; k0 += 64) {
    // Stage A: contiguous 256B rows of h_ws[t].
    for (int idx = tid; idx < 128 * 16; idx += 128) {
      int bl = idx >> 4, k4 = idx & 15;
      v4f v = *(const v4f*)(h_ws + ((size_t)t * BB + mh * 128 + bl) * (TT * HH) + k0 + k4 * 4);
      *(v4f*)(As + bl * ASTRIDE + k4 * 4) = v;
    }
    // Stage B with off-diagonal row remap, into K-pair-interleaved layout.
    for (int idx = tid; idx < 64 * 16; idx += 128) {
      int kl = idx >> 4, e4 = idx & 15;
      int k = k0 + kl;
      int jj = k >> 2;
      int kw = (jj < t) ? k : ((jj > t) ? (k - HH) : 0);
      v4f v = *(const v4f*)(W2 + ((size_t)t * ((TT - 1) * HH) + kw) * EE + e4 * 4);
      float* dst = Bs + (kl >> 1) * BSTRIDE + (kl & 1);
      int e0 = e4 * 4;
      dst[(e0 + 0) * 2] = v.x;
      dst[(e0 + 1) * 2] = v.y;
      dst[(e0 + 2) * 2] = v.z;
      dst[(e0 + 3) * 2] = v.w;
    }
    // Prefetch next A chunk toward L2/WGP$ (lowers to global_prefetch_b8).
    if (k0 + 64 < TT * HH) {
      __builtin_prefetch(h_ws + ((size_t)t * BB + mh * 128 + tid) * (TT * HH) + k0 + 64, 0, 0);
    }
    __syncthreads();

#pragma unroll
    for (int kk = 0; kk < 64; kk += 4) {
      v2f a0 = *(const v2f*)(As + (m0 + lm) * ASTRIDE + kk + half * 2);
      v2f a1 = *(const v2f*)(As + (m0 + 16 + lm) * ASTRIDE + kk + half * 2);
#pragma unroll
      for (int n = 0; n < 4; ++n) {
        v2f bf = *(const v2f*)(Bs + ((kk >> 1) + half) * BSTRIDE + (n * 16 + lm) * 2);
        acc[0][n] = wmma_f32_16x16x4(a0, bf, acc[0][n]);
        acc[1][n] = wmma_f32_16x16x4(a1, bf, acc[1][n]);
      }
    }
    __syncthreads();
  }

#pragma unroll
  for (int mt = 0; mt < 2; ++mt) {
#pragma unroll
    for (int n = 0; n < 4; ++n) {
      int e = n * 16 + lm;
      float bias = b2[(size_t)t * EE + e];
#pragma unroll
      for (int r = 0; r < 8; ++r) {
        int brow = mh * 128 + m0 + mt * 16 + r + half * 8;
        out[((size_t)brow * TT + t) * EE + e] = acc[mt][n][r] + bias;
      }
    }
  }
}

// ---------------------------------------------------------------------------
// Fallback (workspace too small for the 268MB h intermediate): fully fused
// per-(t, b-half) kernel. Stage 1 runs at 25% N-utilization (H=4), but needs
// zero intermediate global traffic. x0 is re-read per t but lives in L2 (17MB).
// ---------------------------------------------------------------------------
__global__ __launch_bounds__(256) void kf_fused(
    const float* __restrict__ x, const float* __restrict__ W1,
    const float* __restrict__ b1, const float* __restrict__ W2,
    const float* __restrict__ b2, float* __restrict__ out) {
  const int t = blockIdx.x;
  const int mh = blockIdx.y;
  const int tid = threadIdx.x;
  const int wave = tid >> 5, lane = tid & 31;
  const int half = lane >> 4, lm = lane & 15;

  __shared__ float Axs[128 * ASTRIDE];  // x0[:,j,:] half-tile
  __shared__ float W1s[EE * 20];        // W1[t,j] as [e][16 cols], cols 4..15 = 0
  __shared__ float W2s[4 * 84];         // 4 remapped W2 rows for this j
  __shared__ float Hb[128 * 4];         // stage-1 activations

  for (int idx = tid; idx < EE * 20; idx += 256) W1s[idx] = 0.f;  // zero pad once
  __syncthreads();

  v8f oacc[4] = {};
  const int m0 = wave * 16;

  for (int j = 0; j < TT; ++j) {
    if (j == t) continue;  // uniform across block
    for (int it = 0; it < 8; ++it) {
      int idx = tid + it * 256;
      int bl = idx >> 4, e4 = idx & 15;
      v4f v = *(const v4f*)(x + (size_t)(mh * 128 + bl) * (TT * EE) + (size_t)j * EE + e4 * 4);
      *(v4f*)(Axs + bl * ASTRIDE + e4 * 4) = v;
    }
    if (tid < 64) {
      int e = tid;
      v4f v = *(const v4f*)(W1 + ((size_t)t * TT + j) * (EE * HH) + e * 4);
      *(v4f*)(W1s + e * 20) = v;
    } else if (tid < 128) {
      int q = tid - 64;
      int kl = q >> 4, e4 = q & 15;
      int jw = (j < t) ? j : (j - 1);
      v4f v = *(const v4f*)(W2 + ((size_t)t * ((TT - 1) * HH) + jw * HH + kl) * EE + e4 * 4);
      *(v4f*)(W2s + kl * 84 + e4 * 4) = v;
    }
    __syncthreads();

    // Stage 1: 16 rows per wave, K=64 -> 16 WMMAs (cols 0..3 valid).
    v8f hacc = {};
    for (int e = 0; e < EE; e += 4) {
      v2f a = *(const v2f*)(Axs + (m0 + lm) * ASTRIDE + e + half * 2);
      v2f bf;
      bf.x = W1s[(e + half * 2 + 0) * 20 + lm];
      bf.y = W1s[(e + half * 2 + 1) * 20 + lm];
      hacc = wmma_f32_16x16x4(a, bf, hacc);
    }
    if (lm < 4) {
      float bias = b1[((size_t)t * TT + j) * HH + lm];
#pragma unroll
      for (int r = 0; r < 8; ++r) {
        float v = hacc[r] + bias;
        Hb[(m0 + r + half * 8) * 4 + lm] = v > 0.f ? v : 0.f;
      }
    }
    __syncthreads();

    // Stage 2: oacc += Hb(16x4) @ W2s(4x64): one K-step, 4 WMMAs.
    {
      v2f a = *(const v2f*)(Hb + (m0 + lm) * 4 + half * 2);
#pragma unroll
      for (int n = 0; n < 4; ++n) {
        v2f bf;
        bf.x = W2s[(half * 2 + 0) * 84 + n * 16 + lm];
        bf.y = W2s[(half * 2 + 1) * 84 + n * 16 + lm];
        oacc[n] = wmma_f32_16x16x4(a, bf, oacc[n]);
      }
    }
    __syncthreads();
  }

#pragma unroll
  for (int n = 0; n < 4; ++n) {
    int e = n * 16 + lm;
    float bias = b2[(size_t)t * EE + e];
#pragma unroll
    for (int r = 0; r < 8; ++r) {
      int brow = mh * 128 + m0 + r + half * 8;
      out[((size_t)brow * TT + t) * EE + e] = oacc[n][r] + bias;
    }
  }
}

extern "C" void kernel_launch(void* const* d_in, const int* in_sizes, int n_in,
                              void* d_out, int out_size, void* d_ws, size_t ws_size,
                              hipStream_t stream) {
  (void)in_sizes; (void)n_in; (void)out_size;
  const float* x  = (const float*)d_in[0];
  const float* W1 = (const float*)d_in[1];
  const float* b1 = (const float*)d_in[2];
  const float* W2 = (const float*)d_in[3];
  const float* b2 = (const float*)d_in[4];
  float* out = (float*)d_out;

  const size_t h_bytes = (size_t)TT * BB * TT * HH * sizeof(float);  // 268 MB
  if (ws_size >= h_bytes) {
    float* h_ws = (float*)d_ws;
    k1_h<<<dim3(TT, TT / 16, 2), 128, 0, stream>>>(x, W1, b1, h_ws);
    k2_out<<<dim3(TT, 2), 128, 0, stream>>>(h_ws, W2, b2, out);
  } else {
    kf_fused<<<dim3(TT, 2), 256, 0, stream>>>(x, W1, b1, W2, b2, out);
  }
}